// RPN_41360535060674
// MI455X (gfx1250) — compile-verified
//
#include <hip/hip_runtime.h>
#include <cstdint>

typedef __attribute__((ext_vector_type(2))) float v2f;
typedef __attribute__((ext_vector_type(8))) float v8f;
typedef __attribute__((ext_vector_type(4))) unsigned int v4u;
typedef __attribute__((ext_vector_type(8))) int v8i;
typedef __attribute__((ext_vector_type(4))) int v4i;

#define CVOL   262144          // 64*64*64 spatial positions
#define KTOT   1728            // 64 ci * 27 taps
#define NANCH  15
#define NSCORE (CVOL * NANCH)  // 3,932,160
#define TOPN   12000
#define SORTN  16384
#define BBOX_CLIP 4.1351665565f   // log(1000/16)

// ---------------------------------------------------------------------------
// Prep kernels:
//  (a) transpose conv weights to tap-major wT[co][tap][ci]  (contiguous A)
//  (b) pack cls(15)+bbox(90) head weights into zero-padded 112x64 matrix
//  (c) transpose x from [ci][p] to position-major xT[p][ci] so a lane's K-pair
//      (adjacent ci) is one aligned global_load_b64 in the conv hot loop.
// ---------------------------------------------------------------------------
__global__ __launch_bounds__(256) void prep_convw_kernel(
    const float* __restrict__ wconv, float* __restrict__ wT) {
  const int idx = blockIdx.x * 256 + threadIdx.x;
  if (idx >= 64 * KTOT) return;
  const int co = idx / KTOT;
  const int r  = idx - co * KTOT;
  const int ci = r & 63;
  const int tap = r >> 6;
  wT[idx] = wconv[co * KTOT + ci * 27 + tap];
}

__global__ __launch_bounds__(256) void prep_headw_kernel(
    const float* __restrict__ wcls, const float* __restrict__ wbb,
    float* __restrict__ wh) {
  const int idx = blockIdx.x * 256 + threadIdx.x;
  if (idx >= 112 * 64) return;
  const int m = idx >> 6;
  const int k = idx & 63;
  float v = 0.0f;
  if (m < 15) v = wcls[m * 64 + k];
  else if (m < 105) v = wbb[(m - 15) * 64 + k];
  wh[idx] = v;
}

__global__ __launch_bounds__(256) void transpose_x_kernel(
    const float* __restrict__ x, float* __restrict__ xT) {
  __shared__ float tb[64][65];
  const int pb = blockIdx.x * 64;          // 4096 blocks of 64 positions
  const int tx = threadIdx.x & 63;
  const int ty = threadIdx.x >> 6;         // 0..3
  for (int c = ty; c < 64; c += 4)
    tb[c][tx] = x[(size_t)c * CVOL + pb + tx];      // coalesced read
  __syncthreads();
  for (int p = ty; p < 64; p += 4)
    xT[(size_t)(pb + p) * 64 + tx] = tb[tx][p];     // coalesced write
}

// ---------------------------------------------------------------------------
// Kernel 1: 3x3x3 conv, SAME padding, bias+ReLU, as implicit GEMM via
// V_WMMA_F32_16X16X4_F32.  Each wave: M=64 (4 co-tiles) x N=16 positions.
// Tap-major K: 27 taps x 16 ci-steps of K=4.  Uniform OOB taps skipped;
// x-edge padding via clamped address + mask multiply (branch-free B path).
// ---------------------------------------------------------------------------
__global__ __launch_bounds__(256) void conv3d_wmma_kernel(
    const float* __restrict__ xT, const float* __restrict__ wT,
    const float* __restrict__ bconv, float* __restrict__ y) {
  const int lane = threadIdx.x & 31;
  const int wave = threadIdx.x >> 5;
  const int tile = blockIdx.x * 8 + wave;   // 16384 N-tiles of 16 positions
  const int p0   = tile * 16;
  const int d  = p0 >> 12;
  const int h  = (p0 >> 6) & 63;
  const int w0 = p0 & 63;
  const int n  = lane & 15;
  const int hi = lane >> 4;
  const int kof = hi << 1;                  // lanes 16-31 hold K+2, K+3
  const int wxn = w0 + n;

  v8f acc0 = (v8f)(0.0f);
  v8f acc1 = (v8f)(0.0f);
  v8f acc2 = (v8f)(0.0f);
  v8f acc3 = (v8f)(0.0f);

  const float* wr0 = wT + (size_t)(n)      * KTOT;
  const float* wr1 = wT + (size_t)(16 + n) * KTOT;
  const float* wr2 = wT + (size_t)(32 + n) * KTOT;
  const float* wr3 = wT + (size_t)(48 + n) * KTOT;

  for (int iz = 0; iz < 3; ++iz) {
    const int zz = d + iz - 1;
    if ((unsigned)zz >= 64u) continue;           // uniform per wave
    for (int iy = 0; iy < 3; ++iy) {
      const int yy = h + iy - 1;
      if ((unsigned)yy >= 64u) continue;         // uniform per wave
      for (int ix = 0; ix < 3; ++ix) {
        const int xx  = wxn + ix - 1;
        const int xxc = min(max(xx, 0), 63);     // clamped: always in-bounds
        const float okf = (xx == xxc) ? 1.0f : 0.0f;   // per-lane edge mask
        const float* xb =
            xT + ((size_t)((zz << 12) + (yy << 6) + xxc) << 6) + kof;
        __builtin_prefetch(xb, 0, 3);
        const int kbase = (((iz * 3 + iy) * 3) + ix) * 64 + kof;
#pragma unroll 4
        for (int ci0 = 0; ci0 < 64; ci0 += 4) {
          const int ka = kbase + ci0;
          const v2f a0 = *(const v2f*)(wr0 + ka);
          const v2f a1 = *(const v2f*)(wr1 + ka);
          const v2f a2 = *(const v2f*)(wr2 + ka);
          const v2f a3 = *(const v2f*)(wr3 + ka);
          const v2f bv = *(const v2f*)(xb + ci0);      // one b64 load
          const v2f b  = bv * okf;                     // zero-fill padding
          acc0 = __builtin_amdgcn_wmma_f32_16x16x4_f32(false, a0, false, b, (short)0, acc0, false, false);
          acc1 = __builtin_amdgcn_wmma_f32_16x16x4_f32(false, a1, false, b, (short)0, acc1, false, false);
          acc2 = __builtin_amdgcn_wmma_f32_16x16x4_f32(false, a2, false, b, (short)0, acc2, false, false);
          acc3 = __builtin_amdgcn_wmma_f32_16x16x4_f32(false, a3, false, b, (short)0, acc3, false, false);
        }
      }
    }
  }

  // C/D layout: VGPR r -> M = r (lanes 0-15) / r+8 (lanes 16-31); N = lane&15
#pragma unroll
  for (int r = 0; r < 8; ++r) {
    const int m = r + 8 * hi;
    { int co = m;       float v = acc0[r] + bconv[co];
      y[(size_t)co * CVOL + p0 + n] = fmaxf(v, 0.0f); }
    { int co = 16 + m;  float v = acc1[r] + bconv[co];
      y[(size_t)co * CVOL + p0 + n] = fmaxf(v, 0.0f); }
    { int co = 32 + m;  float v = acc2[r] + bconv[co];
      y[(size_t)co * CVOL + p0 + n] = fmaxf(v, 0.0f); }
    { int co = 48 + m;  float v = acc3[r] + bconv[co];
      y[(size_t)co * CVOL + p0 + n] = fmaxf(v, 0.0f); }
  }
}

// ---------------------------------------------------------------------------
// Kernel 2: fused heads (packed 112x64 weights) as WMMA GEMM.  The B operand
// (64 rows x 16 floats of y, row stride 1MB) is staged into LDS by the Tensor
// Data Mover (TENSOR_LOAD_TO_LDS + s_wait_tensorcnt), then per-position anchor
// decode / clip / filter -> scores + proposals.
// ---------------------------------------------------------------------------
__global__ __launch_bounds__(256) void heads_wmma_kernel(
    const float* __restrict__ y, const float* __restrict__ wh,
    const float* __restrict__ bcls, const float* __restrict__ bbb,
    float* __restrict__ scores, float* __restrict__ props) {
  __shared__ float ltile[8 * 112 * 16];   // 57344 B; per-wave 7168 B region
  const int lane = threadIdx.x & 31;
  const int wave = threadIdx.x >> 5;
  const int tile = blockIdx.x * 8 + wave;
  const int p0   = tile * 16;
  const int n  = lane & 15;
  const int hi = lane >> 4;
  const int kof = hi << 1;

  float* my = &ltile[wave * 112 * 16];

  // ---- stage B tile (y[k][p0..p0+15], k=0..63) into my[0..1023] ----
#if __has_builtin(__builtin_amdgcn_tensor_load_to_lds)
  {
    unsigned lds_off = (unsigned)(uintptr_t)my;   // low 32 bits = LDS offset
    unsigned long long ga = (unsigned long long)(uintptr_t)(y + p0);
    lds_off = __builtin_amdgcn_readfirstlane(lds_off);
    const unsigned galo = __builtin_amdgcn_readfirstlane((unsigned)ga);
    const unsigned gahi = __builtin_amdgcn_readfirstlane((unsigned)(ga >> 32));
    v4u g0;
    g0.x = 1u;                                    // count=1, user descriptor
    g0.y = lds_off;                               // D#.lds_addr
    g0.z = galo;                                  // D#.global_addr[31:0]
    g0.w = (gahi & 0x1FFFFFFu) | (2u << 30);      // addr[56:32] | type=2
    v8i g1;
    g1[0] = (int)(2u << 16);                      // data_size = 4 bytes
    g1[1] = 0;                                    // tensor_dim0[15:0]=0 (262144)
    g1[2] = (int)(0x4u | (64u << 16));            // dim0 hi16, tensor_dim1=64
    g1[3] = (int)(16u << 16);                     // tile_dim0 = 16
    g1[4] = (int)64u;                             // tile_dim1 = 64, tile_dim2=0
    g1[5] = (int)262144u;                         // tensor_dim0_stride lo32
    g1[6] = 0;
    g1[7] = 0;
    const v4i gz4 = (v4i)0;                       // groups 2/3 unused (2D tile)
    const v8i gz8 = (v8i)0;
    __builtin_amdgcn_tensor_load_to_lds(g0, g1, gz4, gz4, gz8, 0);
    __builtin_amdgcn_s_wait_tensorcnt(0);
  }
  const float* sy = my;
#else
  // fallback: direct staging via plain stores
  for (int k = hi; k < 64; k += 2) my[k * 16 + n] = y[(size_t)k * CVOL + p0 + n];
  __syncthreads();
  const float* sy = my;
#endif

  v8f acc[7];
#pragma unroll
  for (int t = 0; t < 7; ++t) acc[t] = (v8f)(0.0f);

  for (int k0 = 0; k0 < 64; k0 += 4) {
    const int ka = k0 + kof;
    v2f b;
    b.x = sy[ka * 16 + n];
    b.y = sy[(ka + 1) * 16 + n];
#pragma unroll
    for (int t = 0; t < 7; ++t) {
      const v2f a = *(const v2f*)(wh + (size_t)(t * 16 + n) * 64 + ka);
      acc[t] = __builtin_amdgcn_wmma_f32_16x16x4_f32(false, a, false, b, (short)0, acc[t], false, false);
    }
  }

  // write head outputs over the (now consumed) staging region
#pragma unroll
  for (int t = 0; t < 7; ++t)
#pragma unroll
    for (int r = 0; r < 8; ++r)
      my[(t * 16 + r + 8 * hi) * 16 + n] = acc[t][r];
  __syncthreads();

  if (lane < 16) {
    const int p  = p0 + lane;
    const int px = p & 63, py = (p >> 6) & 63, pz = p >> 12;
    const float sx = px * 16.0f, sy2 = py * 16.0f, sz = pz * 16.0f;
    for (int a = 0; a < NANCH; ++a) {
      const int rid = a / 5, sid = a - rid * 5;
      const float sxy   = (float)(2 << sid);
      const float ratio = 0.5f * (float)(1 << rid);
      const float rs    = sqrtf(ratio);
      const float wsA = 16.0f * sxy * rs;
      const float hsA = 16.0f * sxy / rs;
      const float dsA = (float)(8 << sid);
      const float ax0 = 7.5f - 0.5f * (wsA - 1.0f) + sx;
      const float ay0 = 7.5f - 0.5f * (hsA - 1.0f) + sy2;
      const float az0 = 7.5f - 0.5f * (dsA - 1.0f) + sz;
      const float ax1 = 7.5f + 0.5f * (wsA - 1.0f) + sx;
      const float ay1 = 7.5f + 0.5f * (hsA - 1.0f) + sy2;
      const float az1 = 7.5f + 0.5f * (dsA - 1.0f) + sz;
      const float wA = ax1 - ax0 + 1.0f;
      const float hA = ay1 - ay0 + 1.0f;
      const float dA = az1 - az0 + 1.0f;
      const float cx = ax0 + 0.5f * wA;
      const float cy = ay0 + 0.5f * hA;
      const float cz = az0 + 0.5f * dA;

      const float logit = my[a * 16 + lane] + bcls[a];
      const int   db = 15 + a * 6;
      const float dxv = my[(db + 0) * 16 + lane] + bbb[a * 6 + 0];
      const float dyv = my[(db + 1) * 16 + lane] + bbb[a * 6 + 1];
      const float dzv = my[(db + 2) * 16 + lane] + bbb[a * 6 + 2];
      const float dwv = fminf(my[(db + 3) * 16 + lane] + bbb[a * 6 + 3], BBOX_CLIP);
      const float dhv = fminf(my[(db + 4) * 16 + lane] + bbb[a * 6 + 4], BBOX_CLIP);
      const float ddv = fminf(my[(db + 5) * 16 + lane] + bbb[a * 6 + 5], BBOX_CLIP);

      const float pcx = dxv * wA + cx;
      const float pcy = dyv * hA + cy;
      const float pcz = dzv * dA + cz;
      const float pw = expf(dwv) * wA;
      const float ph = expf(dhv) * hA;
      const float pd = expf(ddv) * dA;

      float bx0 = fminf(fmaxf(pcx - 0.5f * pw, 0.0f), 1023.0f);
      float by0 = fminf(fmaxf(pcy - 0.5f * ph, 0.0f), 1023.0f);
      float bz0 = fminf(fmaxf(pcz - 0.5f * pd, 0.0f), 1023.0f);
      float bx1 = fminf(fmaxf(pcx + 0.5f * pw - 1.0f, 0.0f), 1023.0f);
      float by1 = fminf(fmaxf(pcy + 0.5f * ph - 1.0f, 0.0f), 1023.0f);
      float bz1 = fminf(fmaxf(pcz + 0.5f * pd - 1.0f, 0.0f), 1023.0f);

      const float wsz = bx1 - bx0, hsz = by1 - by0, dsz = bz1 - bz0;
      const bool keep = (wsz > 0.0f) && (hsz > 0.0f) && (dsz > 0.0f) &&
                        (bx0 + wsz * 0.5f < 1024.0f) &&
                        (by0 + hsz * 0.5f < 1024.0f) &&
                        (bz0 + dsz * 0.5f < 1024.0f);
      const float sc = keep ? (1.0f / (1.0f + expf(-logit))) : -1.0f;

      const int row = p * NANCH + a;
      scores[row] = sc;
      float* pr = props + (size_t)row * 6;
      pr[0] = bx0; pr[1] = by0; pr[2] = bz0;
      pr[3] = bx1; pr[4] = by1; pr[5] = bz1;
    }
  }
}

// ---------------------------------------------------------------------------
// Top-K = 4-pass radix select + compaction + single-WG bitonic sort + gather.
// ---------------------------------------------------------------------------
__device__ __forceinline__ unsigned score_key(float s) {
  unsigned b = __float_as_uint(s);
  return (b & 0x80000000u) ? ~b : (b | 0x80000000u);
}

__global__ void topk_init_kernel(unsigned* hist, unsigned* state,
                                 unsigned long long* cand) {
  const int t = threadIdx.x;
  for (int i = t; i < 1024; i += 256) hist[i] = 0u;
  if (t == 0) {
    state[0] = 0u;       // prefix
    state[1] = TOPN;     // k remaining within prefix class
    state[2] = 0u;       // final count of keys strictly greater than threshold
    state[4] = 0u;       // atomic counter: greater
    state[5] = 0u;       // atomic counter: equal
  }
  for (int i = t; i < SORTN; i += 256) cand[i] = 0ull;
}

__global__ __launch_bounds__(256) void topk_hist_kernel(
    const float* __restrict__ scores, unsigned* __restrict__ hist,
    const unsigned* __restrict__ state, int pass) {
  __shared__ unsigned lh[256];
  lh[threadIdx.x] = 0u;
  __syncthreads();
  const unsigned prefix = state[0];
  const int total = NSCORE;
  for (int i = blockIdx.x * blockDim.x + threadIdx.x; i < total;
       i += gridDim.x * blockDim.x) {
    const unsigned u = score_key(scores[i]);
    bool match = true;
    if (pass > 0) match = ((u >> (32 - 8 * pass)) == prefix);
    if (match) atomicAdd(&lh[(u >> (24 - 8 * pass)) & 255u], 1u);
  }
  __syncthreads();
  if (lh[threadIdx.x]) atomicAdd(&hist[pass * 256 + threadIdx.x], lh[threadIdx.x]);
}

__global__ void topk_select_kernel(const unsigned* __restrict__ hist,
                                   unsigned* __restrict__ state, int pass) {
  if (threadIdx.x != 0) return;
  const unsigned* h = hist + pass * 256;
  unsigned kRem = state[1];
  unsigned cum = 0u;
  int b = 255;
  for (; b > 0; --b) {
    if (cum + h[b] >= kRem) break;
    cum += h[b];
  }
  state[0] = (state[0] << 8) | (unsigned)b;
  state[1] = kRem - cum;
  if (pass == 3) state[2] = (unsigned)TOPN - state[1];
}

__global__ __launch_bounds__(256) void topk_compact_kernel(
    const float* __restrict__ scores, unsigned* __restrict__ state,
    unsigned long long* __restrict__ cand) {
  const unsigned T = state[0];
  const unsigned cntGT = state[2];
  const int total = NSCORE;
  for (int i = blockIdx.x * blockDim.x + threadIdx.x; i < total;
       i += gridDim.x * blockDim.x) {
    const unsigned u = score_key(scores[i]);
    if (u > T) {
      const unsigned slot = atomicAdd(&state[4], 1u);
      cand[slot] = ((unsigned long long)u << 32) | (unsigned)(~(unsigned)i);
    } else if (u == T) {
      const unsigned s2 = atomicAdd(&state[5], 1u);
      const unsigned slot = cntGT + s2;
      if (slot < TOPN)
        cand[slot] = ((unsigned long long)u << 32) | (unsigned)(~(unsigned)i);
    }
  }
}

__global__ __launch_bounds__(1024) void topk_sort_kernel(
    unsigned long long* __restrict__ cand) {
  extern __shared__ unsigned long long sm[];
  const int t = threadIdx.x;
  for (int i = t; i < SORTN; i += 1024) sm[i] = cand[i];
  __syncthreads();
  for (unsigned k = 2; k <= SORTN; k <<= 1) {
    for (unsigned j = k >> 1; j > 0; j >>= 1) {
      for (unsigned i = t; i < SORTN; i += 1024) {
        const unsigned ixj = i ^ j;
        if (ixj > i) {
          const bool desc = ((i & k) == 0);
          unsigned long long va = sm[i], vb = sm[ixj];
          if (desc ? (va < vb) : (va > vb)) { sm[i] = vb; sm[ixj] = va; }
        }
      }
      __syncthreads();
    }
  }
  for (int i = t; i < SORTN; i += 1024) cand[i] = sm[i];
}

__global__ __launch_bounds__(256) void topk_gather_kernel(
    const unsigned long long* __restrict__ cand,
    const float* __restrict__ scores, const float* __restrict__ props,
    float* __restrict__ out) {
  const int i = blockIdx.x * blockDim.x + threadIdx.x;
  if (i >= TOPN) return;
  const unsigned long long e = cand[i];
  const unsigned idx = ~(unsigned)(e & 0xFFFFFFFFull);
  const float* pr = props + (size_t)idx * 6;
  float* o = out + (size_t)i * 7;
  o[0] = pr[0]; o[1] = pr[1]; o[2] = pr[2];
  o[3] = pr[3]; o[4] = pr[4]; o[5] = pr[5];
  o[6] = scores[idx];
}

// ---------------------------------------------------------------------------
extern "C" void kernel_launch(void* const* d_in, const int* in_sizes, int n_in,
                              void* d_out, int out_size, void* d_ws, size_t ws_size,
                              hipStream_t stream) {
  const float* x     = (const float*)d_in[0];
  const float* wconv = (const float*)d_in[1];
  const float* bconv = (const float*)d_in[2];
  const float* wcls  = (const float*)d_in[3];
  const float* bcls  = (const float*)d_in[4];
  const float* wbb   = (const float*)d_in[5];
  const float* bbb   = (const float*)d_in[6];
  float* out = (float*)d_out;

  // workspace carve-up
  float* y      = (float*)d_ws;                       // 64 * 262144 floats
  float* xT     = y + (size_t)64 * CVOL;              // 64 * 262144 floats
  float* scores = xT + (size_t)64 * CVOL;             // NSCORE floats
  float* props  = scores + (size_t)NSCORE;            // NSCORE * 6 floats
  float* wT     = props + (size_t)NSCORE * 6;         // 110592 floats
  float* wh     = wT + (size_t)64 * KTOT;             // 7168 floats
  unsigned* hist  = (unsigned*)(wh + 112 * 64);       // 1024
  unsigned* state = hist + 1024;                      // 16
  uintptr_t candAddr = ((uintptr_t)(state + 16) + 255) & ~(uintptr_t)255;
  unsigned long long* cand = (unsigned long long*)candAddr;    // SORTN u64

  // 0) data re-layout
  prep_convw_kernel<<<(64 * KTOT + 255) / 256, 256, 0, stream>>>(wconv, wT);
  prep_headw_kernel<<<(112 * 64 + 255) / 256, 256, 0, stream>>>(wcls, wbb, wh);
  transpose_x_kernel<<<CVOL / 64, 256, 0, stream>>>(x, xT);
  // 1) conv (implicit GEMM, fp32 WMMA, tap-major K, position-major x)
  conv3d_wmma_kernel<<<2048, 256, 0, stream>>>(xT, wT, bconv, y);
  // 2) fused heads (TDM-staged B) + proposal decode
  heads_wmma_kernel<<<2048, 256, 0, stream>>>(y, wh, bcls, bbb, scores, props);
  // 3) radix-select top-12000
  topk_init_kernel<<<1, 256, 0, stream>>>(hist, state, cand);
  for (int pass = 0; pass < 4; ++pass) {
    topk_hist_kernel<<<1024, 256, 0, stream>>>(scores, hist, state, pass);
    topk_select_kernel<<<1, 32, 0, stream>>>(hist, state, pass);
  }
  topk_compact_kernel<<<1024, 256, 0, stream>>>(scores, state, cand);
  topk_sort_kernel<<<1, 1024, SORTN * sizeof(unsigned long long), stream>>>(cand);
  topk_gather_kernel<<<(TOPN + 255) / 256, 256, 0, stream>>>(cand, scores,
                                                             props, out);
}